// MCPBRNN_PETconstraint_MassRelax_Regular_53695681134743
// MI455X (gfx1250) — compile-verified
//
#include <hip/hip_runtime.h>
#include <hip/hip_bf16.h>
#include <math.h>

// ---------------- problem constants (from reference) ----------------
#define BB        100000          // scan length (== "B" in reference)
#define SPIN_LEN  365
#define TRAIN_LEN 80000
#define NRED      (TRAIN_LEN - SPIN_LEN)   // 79635 elements in std()
#define MLc       2.9086f
#define SLc       1.898f
#define SCALE_MR  500.0f
#define NBLK_RED  64              // reduction waves (1 wave per block)

// ---------------- gfx1250 hw transcendentals (guarded) ----------------
#if defined(__has_builtin)
#  if __has_builtin(__builtin_amdgcn_tanh_f32)
#    define TANHF(x) __builtin_amdgcn_tanh_f32(x)
#  endif
#  if __has_builtin(__builtin_amdgcn_rcpf)
#    define RCPF(x) __builtin_amdgcn_rcpf(x)
#  endif
#endif
#ifndef TANHF
#  define TANHF(x) tanhf(x)
#endif
#ifndef RCPF
#  define RCPF(x) (1.0f / (x))
#endif

// sigmoid(z) = 0.5 + 0.5*tanh(0.5*z)  -> fma, tanh, fma
__device__ __forceinline__ float sigmoid_t(float z) {
    return fmaf(TANHF(0.5f * z), 0.5f, 0.5f);
}

// broadcast lane i (SGPR or imm lane select): v_readlane_b32, no LDS
__device__ __forceinline__ float readlane_f(float v, int lane) {
    return __builtin_bit_cast(float,
        __builtin_amdgcn_readlane(__builtin_bit_cast(int, v), lane));
}

typedef float v2f __attribute__((ext_vector_type(2)));
typedef float v8f __attribute__((ext_vector_type(8)));

#if defined(__has_builtin)
#  if __has_builtin(__builtin_amdgcn_wmma_f32_16x16x4_f32)
#    define HAVE_WMMA4 1
#  endif
#endif
#ifndef HAVE_WMMA4
#  define HAVE_WMMA4 0
#endif

// =====================================================================
// Kernel 1: parallel precompute of ol[t] (depends only on u2[t]) and
//           zero-fill of outputs 4 and 5.
// =====================================================================
__global__ __launch_bounds__(256)
void precompute_ol_kernel(const float* __restrict__ x,
                          const float* __restrict__ w_r_yom,
                          const float* __restrict__ w_r_ylm,
                          const float* __restrict__ w_r_yfm,
                          const float* __restrict__ b0_ylm,
                          const float* __restrict__ w2_ylm,
                          float* __restrict__ olbuf,
                          float* __restrict__ out)
{
    int t = blockIdx.x * blockDim.x + threadIdx.x;
    if (t >= BB) return;
    float eo = __expf(w_r_yom[0]);
    float el = __expf(w_r_ylm[0]);
    float ef = __expf(w_r_yfm[0]);
    float ol1 = el / (eo + el + ef);
    float b   = b0_ylm[0];
    float w2  = w2_ylm[0];

    float u2 = x[2 * t + 1];
    float z  = fmaf((u2 - MLc) * (1.0f / SLc), w2, b);
    olbuf[t] = ol1 * sigmoid_t(z);

    __builtin_nontemporal_store(0.0f, &out[4 * BB + t]);   // zeros #5
    __builtin_nontemporal_store(0.0f, &out[5 * BB + t]);   // zeros #6
}

// =====================================================================
// Kernel 2: per-wave partial (sum, sumsq) of y_obs[SPIN_LEN:TRAIN_LEN)
//           via V_WMMA_F32_16X16X4_F32:  D = A(16x4) * ones(4x16) + C.
//           Sum of all 256 C entries == 16 * total.
// =====================================================================
__global__ __launch_bounds__(32)
void reduce_partial_kernel(const float* __restrict__ y,
                           float* __restrict__ part)
{
    const int lane = threadIdx.x;
    const int wave = blockIdx.x;

    v8f acc   = {};
    v8f accsq = {};
    v2f ones; ones.x = 1.0f; ones.y = 1.0f;

    for (int base = SPIN_LEN + wave * 64; base < TRAIN_LEN; base += NBLK_RED * 64) {
        float e0, e1;
        if (base + 64 <= TRAIN_LEN) {          // wave-uniform fast path
            e0 = y[base + 2 * lane];
            e1 = y[base + 2 * lane + 1];
        } else {                               // masked tail (last chunk only)
            int p0 = base + 2 * lane;
            int p1 = p0 + 1;
            e0 = (p0 < TRAIN_LEN) ? y[p0] : 0.0f;
            e1 = (p1 < TRAIN_LEN) ? y[p1] : 0.0f;
        }
#if HAVE_WMMA4
        v2f a;  a.x  = e0;      a.y  = e1;
        v2f a2; a2.x = e0 * e0; a2.y = e1 * e1;
        acc   = __builtin_amdgcn_wmma_f32_16x16x4_f32(false, a,  false, ones,
                                                      (short)0, acc,   false, false);
        accsq = __builtin_amdgcn_wmma_f32_16x16x4_f32(false, a2, false, ones,
                                                      (short)0, accsq, false, false);
#else
        acc[0]   += e0 + e1;
        accsq[0] += e0 * e0 + e1 * e1;
#endif
    }

    float s = 0.0f, q = 0.0f;
    #pragma unroll
    for (int i = 0; i < 8; ++i) { s += acc[i]; q += accsq[i]; }
    #pragma unroll
    for (int o = 16; o > 0; o >>= 1) {
        s += __shfl_xor(s, o, 32);
        q += __shfl_xor(q, o, 32);
    }
#if HAVE_WMMA4
    const float DIV = 16.0f;   // 16 identical columns in C/D
#else
    const float DIV = 1.0f;
#endif
    if (lane == 0) {
        part[2 * wave + 0] = s / DIV;
        part[2 * wave + 1] = q / DIV;
    }
}

// =====================================================================
// Kernel 3: final std with ddof=1 from partials.
// =====================================================================
__global__ __launch_bounds__(32)
void reduce_final_kernel(const float* __restrict__ part,
                         float* __restrict__ stdp)
{
    if (threadIdx.x == 0 && blockIdx.x == 0) {
        float s = 0.0f, q = 0.0f;
        for (int i = 0; i < NBLK_RED; ++i) {
            s += part[2 * i + 0];
            q += part[2 * i + 1];
        }
        float n   = (float)NRED;
        float var = (q - s * s / n) / (n - 1.0f);
        stdp[0] = sqrtf(fmaxf(var, 0.0f));
    }
}

// =====================================================================
// Kernel 4: the inherently-serial 100000-step recurrence. One wave32.
//  - software-pipelined batch loads + global_prefetch_b8 ahead
//  - v_readlane_b32 broadcasts; unroll-by-8 keeps loop I$-resident
//    (only one wave on the chip: an I$ miss stalls the critical path)
//  - hw tanh + hw rcp; c1 tail fused into one extra FMA
//  - per-lane retain of "my" step, nontemporal coalesced stores
// =====================================================================
__global__ __launch_bounds__(32)
void scan_kernel(const float*  __restrict__ x,
                 const float*  __restrict__ olbuf,
                 const float*  __restrict__ stdp,
                 const float*  __restrict__ p_mean,
                 const float*  __restrict__ p_std,
                 const float*  __restrict__ w_r_yom,
                 const float*  __restrict__ w_r_ylm,
                 const float*  __restrict__ w_r_yfm,
                 const float*  __restrict__ w_r_yvm,
                 const float*  __restrict__ b0_yom,
                 const float*  __restrict__ w1_yom,
                 const float*  __restrict__ w_s_yvm,
                 const float*  __restrict__ b0_yrm,
                 float* __restrict__ out)
{
    const int lane = threadIdx.x;

    // ---- wave-uniform scalar parameters / derived constants ----
    float eo = __expf(w_r_yom[0]);
    float el = __expf(w_r_ylm[0]);
    float ef = __expf(w_r_yfm[0]);
    float oo1 = eo / (eo + el + ef);

    float mo = p_mean[0];
    float so = p_std[0];
    float b_yom = b0_yom[0];
    float w1    = w1_yom[0];
    float wsx   = __expf(w_s_yvm[0]);
    float sv    = sigmoid_t(w_r_yvm[0]);
    float ebr   = __expf(b0_yrm[0]);
    float obsstd = stdp[0];

    // sigmoid(b_yom + (c0-mo)/so*w1) as 0.5+0.5*tanh(zs*c0+zc)
    float k1 = w1 / so;
    float zs = 0.5f * k1;
    float zc = 0.5f * (b_yom - mo * k1);
    float ooA = 0.5f * oo1;             // oo = ooA*tanh + ooA
    // tanh((c0/SCALE - ebr)*ws) = tanh(wsc*c0 - ebw)
    float wsc  = wsx * (1.0f / SCALE_MR);
    float ebw  = ebr * wsx;
    float ebrS = ebr * SCALE_MR;

    const v2f* xp = (const v2f*)x;

    // pipeline prologue: batch 0 in registers
    v2f   uv  = xp[lane];
    float olv = olbuf[lane];

    float c0 = 0.0f;
    for (int b = 0; b < BB; b += 32) {
        // issue next batch's loads now; first use is ~32 steps away
        int nb = (b + 32 < BB) ? (b + 32) : b;
        v2f   uv_n  = xp[nb + lane];
        float olv_n = olbuf[nb + lane];

        // prefetch ~4 batches ahead (speculative; OOB dropped by HW)
        __builtin_prefetch(&xp[b + 128], 0, 0);
        __builtin_prefetch(&olbuf[b + 128], 0, 0);

        float kh = 0.f, kc = 0.f, kl = 0.f, klc = 0.f, koo = 0.f,
              kol = 0.f, kolc = 0.f, kf = 0.f, kov = 0.f, kmr = 0.f;

        #pragma unroll 8
        for (int i = 0; i < 32; ++i) {
            float u1t = readlane_f(uv.x, i);
            float u2t = readlane_f(uv.y, i);
            float olt = readlane_f(olv, i);

            // off-chain helper (depends only on c0):
            float absA = fabsf(c0 - ebrS);

            // critical chain (wave-uniform):
            float oo  = fmaf(TANHF(fmaf(c0, zs, zc)), ooA, ooA);
            bool  pos = (c0 > 0.0f);
            float q   = u2t * RCPF(pos ? c0 : 1.0f);
            float olc = pos ? fminf(olt, q) : olt;   // ol - relu(ol - q)
            float f   = (1.0f - oo) - olc;
            float ov1 = sv * TANHF(fmaf(c0, wsc, -ebw));
            float ov  = fminf(ov1, f);               // ov1 - relu(ov1 - f)
            float base = fmaf(f, c0, u1t);           // parallel with ov
            float c1  = fmaf(-ov, absA, base);       // == base - mr
            float mr  = ov * absA;                   // output only (off-chain)

            bool k = (lane == i);
            kh   = k ? oo  * c0 : kh;
            kc   = k ? c0       : kc;
            kl   = k ? olt * c0 : kl;
            klc  = k ? olc * c0 : klc;
            koo  = k ? oo       : koo;
            kol  = k ? olt      : kol;
            kolc = k ? olc      : kolc;
            kf   = k ? f        : kf;
            kov  = k ? ov       : kov;
            kmr  = k ? mr       : kmr;

            c0 = c1;
        }

        int t = b + lane;
        __builtin_nontemporal_store(kh,  &out[ 0 * BB + t]);   // h_n
        __builtin_nontemporal_store(kc,  &out[ 1 * BB + t]);   // c_n
        __builtin_nontemporal_store(kl,  &out[ 2 * BB + t]);   // l_n
        __builtin_nontemporal_store(klc, &out[ 3 * BB + t]);   // lc_n
        __builtin_nontemporal_store(koo, &out[ 6 * BB + t]);   // g_oo
        __builtin_nontemporal_store(kol, &out[ 7 * BB + t]);   // g_ol
        __builtin_nontemporal_store(kolc,&out[ 8 * BB + t]);   // g_olc
        __builtin_nontemporal_store(kf,  &out[ 9 * BB + t]);   // g_f
        v2f hn2; hn2.x = kh; hn2.y = obsstd;                   // h_nout row
        __builtin_nontemporal_store(hn2, (v2f*)(out + 10 * BB) + t);
        __builtin_nontemporal_store(obsstd, &out[12 * BB + t]);// obs_std
        __builtin_nontemporal_store(kov, &out[13 * BB + t]);   // g_ov
        __builtin_nontemporal_store(kmr, &out[14 * BB + t]);   // mr_n

        uv  = uv_n;                   // rotate pipeline
        olv = olv_n;
    }
}

// =====================================================================
// launch
// =====================================================================
extern "C" void kernel_launch(void* const* d_in, const int* in_sizes, int n_in,
                              void* d_out, int out_size, void* d_ws, size_t ws_size,
                              hipStream_t stream)
{
    (void)in_sizes; (void)n_in; (void)out_size; (void)ws_size;

    const float* x        = (const float*)d_in[0];
    // d_in[1] epoch, d_in[2] time_lag : unused (as in reference)
    const float* y_obs    = (const float*)d_in[3];
    const float* p_mean   = (const float*)d_in[4];
    const float* p_std    = (const float*)d_in[5];
    const float* w_r_yom  = (const float*)d_in[6];
    const float* w_r_ylm  = (const float*)d_in[7];
    const float* w_r_yfm  = (const float*)d_in[8];
    const float* w_r_yvm  = (const float*)d_in[9];
    const float* b0_yom   = (const float*)d_in[10];
    const float* w1_yom   = (const float*)d_in[11];
    const float* b0_ylm   = (const float*)d_in[12];
    const float* w2_ylm   = (const float*)d_in[13];
    const float* w_s_yvm  = (const float*)d_in[14];
    const float* b0_yrm   = (const float*)d_in[15];

    float* out  = (float*)d_out;
    float* wsf  = (float*)d_ws;
    float* olbuf = wsf;                       // BB floats
    float* part  = wsf + BB;                  // 2*NBLK_RED floats
    float* stdp  = wsf + BB + 2 * NBLK_RED;   // 1 float

    precompute_ol_kernel<<<(BB + 255) / 256, 256, 0, stream>>>(
        x, w_r_yom, w_r_ylm, w_r_yfm, b0_ylm, w2_ylm, olbuf, out);

    reduce_partial_kernel<<<NBLK_RED, 32, 0, stream>>>(y_obs, part);

    reduce_final_kernel<<<1, 32, 0, stream>>>(part, stdp);

    scan_kernel<<<1, 32, 0, stream>>>(
        x, olbuf, stdp, p_mean, p_std, w_r_yom, w_r_ylm, w_r_yfm, w_r_yvm,
        b0_yom, w1_yom, w_s_yvm, b0_yrm, out);
}